// TropicalLinear_17411797418542
// MI455X (gfx1250) — compile-verified
//
#include <hip/hip_runtime.h>
#include <cstdint>
#include <cstddef>

// Tropical (max-plus) matmul + bias + fused LayerNorm for MI455X (gfx1250).
// VALU-bound (9.7 Gops f32 vs ~50MB HBM traffic). Key points:
//  * double-buffered async global->LDS staging of W k-chunks (ASYNCcnt path),
//    GVS addressing: one SGPR base + one voffset VGPR + one LDS-dest VGPR,
//    6 loads per chunk distinguished only by offset: (applies to BOTH sides)
//  * v_max3_f32-friendly inner loop (acc = max(acc, max(t0,t1)))
//  * 2 B-rows per lane reusing each ds_load_b128 of W
//  * fused LayerNorm epilogue with wave32 __shfl_xor reductions

static constexpr int Bsz = 8192;
static constexpr int Kdim = 768;
static constexpr int Odim = 768;
static constexpr int KC = 8;            // k-chunk rows of W staged per buffer
static constexpr int NC = Kdim / KC;    // 96 chunks
static constexpr int ROWS = 16;         // B-rows per block (2 per wave)
static constexpr int THREADS = 256;     // 8 wave32
static constexpr float EPS = 1e-5f;
static constexpr int CHUNK_BYTES = KC * Odim * 4;   // 24576

struct alignas(16) F4 { float x, y, z, w; };

// One chunk = 6 async b128 per thread. saddr = chunk base (SGPR pair),
// voff = t*16 (VGPR), dst = LDS byte address of this thread's first b128.
// INST_OFFSET advances global source AND LDS dest together (ISA 08 §4.4).
__device__ __forceinline__ void issue_chunk(uint32_t dst, uint32_t voff,
                                            const float* sbase) {
    asm volatile(
        "global_load_async_to_lds_b128 %0, %1, %2 offset:0\n\t"
        "global_load_async_to_lds_b128 %0, %1, %2 offset:4096\n\t"
        "global_load_async_to_lds_b128 %0, %1, %2 offset:8192\n\t"
        "global_load_async_to_lds_b128 %0, %1, %2 offset:12288\n\t"
        "global_load_async_to_lds_b128 %0, %1, %2 offset:16384\n\t"
        "global_load_async_to_lds_b128 %0, %1, %2 offset:20480"
        :: "v"(dst), "v"(voff), "s"(sbase) : "memory");
}

__global__ __launch_bounds__(THREADS) void tropical_ln_kernel(
    const float* __restrict__ x, const float* __restrict__ w,
    const float* __restrict__ bias, const float* __restrict__ gamma,
    const float* __restrict__ beta, float* __restrict__ out) {

    __shared__ alignas(16) float wsh[2][KC * Odim];   // 2 x 24 KB

    const int t    = threadIdx.x;
    const int lane = t & 31;
    const int wave = t >> 5;
    const int r0   = blockIdx.x * ROWS + wave * 2;

    // Low 32 bits of a generic LDS pointer == byte offset within LDS aperture.
    const uint32_t lds_base = (uint32_t)(uintptr_t)(&wsh[0][0]);
    const uint32_t voff = (uint32_t)t * 16u;          // per-lane source offset
    const uint32_t dst0 = lds_base + voff;            // buffer 0 dest
    const uint32_t dst1 = dst0 + (uint32_t)CHUNK_BYTES;

    issue_chunk(dst0, voff, w);                       // chunk 0
    issue_chunk(dst1, voff, w + KC * Odim);           // chunk 1

    F4 acc0[6], acc1[6];
    #pragma unroll
    for (int j = 0; j < 6; ++j) {
        acc0[j].x = acc0[j].y = acc0[j].z = acc0[j].w = -__builtin_huge_valf();
        acc1[j].x = acc1[j].y = acc1[j].z = acc1[j].w = -__builtin_huge_valf();
    }

    for (int c = 0; c < NC; ++c) {
        if (c + 1 < NC) asm volatile("s_wait_asynccnt 6" ::: "memory");
        else            asm volatile("s_wait_asynccnt 0" ::: "memory");
        __syncthreads();                       // chunk c visible to all waves

        const float* wb = wsh[c & 1];

        // wave-uniform x values for this chunk (broadcast b128 loads)
        const F4* xr0 = reinterpret_cast<const F4*>(x + (size_t)r0 * Kdim + c * KC);
        const F4* xr1 = reinterpret_cast<const F4*>(x + (size_t)(r0 + 1) * Kdim + c * KC);
        const F4 xa = xr0[0], xb = xr0[1];
        const F4 xc = xr1[0], xd = xr1[1];
        const float xs0[8] = {xa.x, xa.y, xa.z, xa.w, xb.x, xb.y, xb.z, xb.w};
        const float xs1[8] = {xc.x, xc.y, xc.z, xc.w, xd.x, xd.y, xd.z, xd.w};

        #pragma unroll
        for (int kk = 0; kk < KC; kk += 2) {
            #pragma unroll
            for (int j = 0; j < 6; ++j) {
                const int col = lane * 4 + 128 * j;
                const F4 w0 = *reinterpret_cast<const F4*>(&wb[kk * Odim + col]);
                const F4 w1 = *reinterpret_cast<const F4*>(&wb[(kk + 1) * Odim + col]);
                const float a0 = xs0[kk], a1 = xs0[kk + 1];
                const float b0 = xs1[kk], b1 = xs1[kk + 1];
                // acc = max(acc, max(t0, t1))  ->  v_max3_f32
                acc0[j].x = fmaxf(acc0[j].x, fmaxf(a0 + w0.x, a1 + w1.x));
                acc0[j].y = fmaxf(acc0[j].y, fmaxf(a0 + w0.y, a1 + w1.y));
                acc0[j].z = fmaxf(acc0[j].z, fmaxf(a0 + w0.z, a1 + w1.z));
                acc0[j].w = fmaxf(acc0[j].w, fmaxf(a0 + w0.w, a1 + w1.w));
                acc1[j].x = fmaxf(acc1[j].x, fmaxf(b0 + w0.x, b1 + w1.x));
                acc1[j].y = fmaxf(acc1[j].y, fmaxf(b0 + w0.y, b1 + w1.y));
                acc1[j].z = fmaxf(acc1[j].z, fmaxf(b0 + w0.z, b1 + w1.z));
                acc1[j].w = fmaxf(acc1[j].w, fmaxf(b0 + w0.w, b1 + w1.w));
            }
        }
        __syncthreads();                       // everyone done reading buf c&1
        if (c + 2 < NC) {                      // overwrite it with chunk c+2
            issue_chunk((c & 1) ? dst1 : dst0, voff,
                        w + (size_t)(c + 2) * (KC * Odim));
        }
    }

    // ---- fused bias + LayerNorm epilogue ----
    float s0 = 0.f, q0 = 0.f, s1 = 0.f, q1 = 0.f;
    #pragma unroll
    for (int j = 0; j < 6; ++j) {
        const F4 bb = reinterpret_cast<const F4*>(bias)[lane + 32 * j];
        acc0[j].x += bb.x; acc0[j].y += bb.y; acc0[j].z += bb.z; acc0[j].w += bb.w;
        acc1[j].x += bb.x; acc1[j].y += bb.y; acc1[j].z += bb.z; acc1[j].w += bb.w;
        s0 += acc0[j].x + acc0[j].y + acc0[j].z + acc0[j].w;
        q0 += acc0[j].x * acc0[j].x + acc0[j].y * acc0[j].y
            + acc0[j].z * acc0[j].z + acc0[j].w * acc0[j].w;
        s1 += acc1[j].x + acc1[j].y + acc1[j].z + acc1[j].w;
        q1 += acc1[j].x * acc1[j].x + acc1[j].y * acc1[j].y
            + acc1[j].z * acc1[j].z + acc1[j].w * acc1[j].w;
    }
    #pragma unroll
    for (int off = 16; off > 0; off >>= 1) {
        s0 += __shfl_xor(s0, off, 32);
        q0 += __shfl_xor(q0, off, 32);
        s1 += __shfl_xor(s1, off, 32);
        q1 += __shfl_xor(q1, off, 32);
    }
    const float rn  = 1.0f / (float)Odim;
    const float mu0 = s0 * rn, mu1 = s1 * rn;
    const float inv0 = rsqrtf(q0 * rn - mu0 * mu0 + EPS);
    const float inv1 = rsqrtf(q1 * rn - mu1 * mu1 + EPS);

    float* o0 = out + (size_t)r0 * Odim;
    float* o1 = o0 + Odim;
    #pragma unroll
    for (int j = 0; j < 6; ++j) {
        const F4 g  = reinterpret_cast<const F4*>(gamma)[lane + 32 * j];
        const F4 be = reinterpret_cast<const F4*>(beta)[lane + 32 * j];
        F4 r;
        r.x = (acc0[j].x - mu0) * inv0 * g.x + be.x;
        r.y = (acc0[j].y - mu0) * inv0 * g.y + be.y;
        r.z = (acc0[j].z - mu0) * inv0 * g.z + be.z;
        r.w = (acc0[j].w - mu0) * inv0 * g.w + be.w;
        reinterpret_cast<F4*>(o0)[lane + 32 * j] = r;
        r.x = (acc1[j].x - mu1) * inv1 * g.x + be.x;
        r.y = (acc1[j].y - mu1) * inv1 * g.y + be.y;
        r.z = (acc1[j].z - mu1) * inv1 * g.z + be.z;
        r.w = (acc1[j].w - mu1) * inv1 * g.w + be.w;
        reinterpret_cast<F4*>(o1)[lane + 32 * j] = r;
    }
}

extern "C" void kernel_launch(void* const* d_in, const int* in_sizes, int n_in,
                              void* d_out, int out_size, void* d_ws, size_t ws_size,
                              hipStream_t stream) {
    (void)in_sizes; (void)n_in; (void)out_size; (void)d_ws; (void)ws_size;
    const float* x      = (const float*)d_in[0];
    const float* weight = (const float*)d_in[1];
    const float* bias   = (const float*)d_in[2];
    const float* gamma  = (const float*)d_in[3];
    const float* beta   = (const float*)d_in[4];
    float* out = (float*)d_out;

    dim3 grid(Bsz / ROWS);   // 512 blocks
    dim3 block(THREADS);     // 8 wave32
    tropical_ln_kernel<<<grid, block, 0, stream>>>(x, weight, bias, gamma, beta, out);
}